// WaveNetModel_20710332301586
// MI455X (gfx1250) — compile-verified
//
#include <hip/hip_runtime.h>

typedef float v2f __attribute__((ext_vector_type(2)));
typedef float v8f __attribute__((ext_vector_type(8)));

#define NL 40   // total layers (BLOCKS*LAYERS)

// D = A(16x4 f32) * B(4x16 f32) + C(16x16 f32)
__device__ __forceinline__ v8f wmma_k4(v2f a, v2f b, v8f c) {
  return __builtin_amdgcn_wmma_f32_16x16x4_f32(false, a, false, b, (short)0, c,
                                               false, false);
}

__device__ __forceinline__ float fast_tanhf(float x) {
#if __has_builtin(__builtin_amdgcn_tanhf)
  return __builtin_amdgcn_tanhf(x);       // gfx1250 V_TANH_F32 (codegen-confirmed)
#else
  const float e = __expf(-2.0f * fabsf(x));
  return copysignf((1.0f - e) / (1.0f + e), x);
#endif
}

__device__ __forceinline__ float fast_sigmoidf(float x) {
  // v_exp_f32 + v_rcp_f32: avoid the IEEE div_scale/div_fmas Newton sequence.
  return __builtin_amdgcn_rcpf(1.0f + __expf(-x));
}

__global__ __launch_bounds__(256) void wavenet_last_kernel(
    const float* __restrict__ x,        // [8,6,8192]
    const float* __restrict__ start_w,  // [32,6,1]
    const float* __restrict__ filter_w, // [40,32,32,2]
    const float* __restrict__ gate_w,   // [40,32,32,2]
    const float* __restrict__ res_w,    // [40,32,32,1]
    const float* __restrict__ skip_w,   // [40,256,32,1]
    const float* __restrict__ end1_w,   // [256,256,1]
    const float* __restrict__ end1_b,   // [256]
    const float* __restrict__ end2_w,   // [1,256,1]
    const float* __restrict__ end2_b,   // [1]
    float* __restrict__ out)            // [8]
{
  // LDS state. B-matrix layout convention: M[k*ncols + n], batch n in cols 0..7.
  __shared__ float Vl[32 * 16];       // residual state, padded to 16 cols (cols 8..15 = 0)
  __shared__ float Ul[32 * 16];       // gated activation of current layer
  __shared__ float Uh[NL * 32 * 8];   // history of U_i for the skip GEMM
  __shared__ float Sl[256 * 8];       // relu(skip_sum)
  __shared__ float Hl[256 * 8];       // relu(end1)

  const int tid  = threadIdx.x;
  const int wave = tid >> 5;
  const int lane = tid & 31;
  const int row  = lane & 15;  // M-row for A loads, N-col for B/C/D
  const int half = lane >> 4;

  // ---- Phase 0: V = start_w @ x[:, :, -1]  (32 x 8, zero-padded to 16 cols) ----
  for (int idx = tid; idx < 32 * 16; idx += 256) {
    const int c = idx >> 4, n = idx & 15;
    float v = 0.0f;
    if (n < 8) {
      #pragma unroll
      for (int ci = 0; ci < 6; ++ci)
        v += start_w[c * 6 + ci] * x[(n * 6 + ci) * 8192 + 8191];
    }
    Vl[idx] = v;
  }
  __syncthreads();

  // ---- Phase 1: sequential 40-layer recurrence (wave 0, EXEC all-ones) ----
  if (wave == 0) {
    #pragma unroll 1
    for (int i = 0; i < NL; ++i) {
      const float* Wf = filter_w + i * (32 * 32 * 2);  // only tap 0 reaches t=-1
      const float* Wg = gate_w   + i * (32 * 32 * 2);
      const float* Wr = res_w    + i * (32 * 32);

      // Software-pipeline: pull layer i+1's weights to WGP scope while computing i.
      if (i + 1 < NL) {
        const float* nWf = filter_w + (i + 1) * (32 * 32 * 2);
        const float* nWg = gate_w   + (i + 1) * (32 * 32 * 2);
        const float* nWr = res_w    + (i + 1) * (32 * 32);
        __builtin_prefetch(nWf + lane * 64, 0, 3);       // 8KB block, 128B/lane step
        __builtin_prefetch(nWf + lane * 64 + 32, 0, 3);
        __builtin_prefetch(nWg + lane * 64, 0, 3);
        __builtin_prefetch(nWg + lane * 64 + 32, 0, 3);
        __builtin_prefetch(nWr + lane * 32, 0, 3);       // 4KB block
      }

      // F = Wf0 @ V ; G = Wg0 @ V ; U = tanh(F) * sigmoid(G)
      #pragma unroll 1
      for (int m = 0; m < 2; ++m) {
        v8f fa = {};
        v8f ga = {};
        #pragma unroll
        for (int k0 = 0; k0 < 32; k0 += 4) {
          const int ka = k0 + 2 * half;
          v2f af, ag, bv;
          af.x = Wf[((m * 16 + row) * 32 + ka) * 2];      // tap 0
          af.y = Wf[((m * 16 + row) * 32 + ka + 1) * 2];
          ag.x = Wg[((m * 16 + row) * 32 + ka) * 2];
          ag.y = Wg[((m * 16 + row) * 32 + ka + 1) * 2];
          bv.x = Vl[(ka + 0) * 16 + row];
          bv.y = Vl[(ka + 1) * 16 + row];
          fa = wmma_k4(af, bv, fa);
          ga = wmma_k4(ag, bv, ga);
        }
        #pragma unroll
        for (int r = 0; r < 8; ++r) {
          const int c = m * 16 + r + 8 * half;
          const float u = fast_tanhf(fa[r]) * fast_sigmoidf(ga[r]);
          Ul[c * 16 + row] = u;                       // padded cols stay exactly 0
          if (row < 8) Uh[(i * 32 + c) * 8 + row] = u;
        }
      }

      // V' = Wr @ U + U   (residual)
      #pragma unroll 1
      for (int m = 0; m < 2; ++m) {
        v8f ra = {};
        #pragma unroll
        for (int k0 = 0; k0 < 32; k0 += 4) {
          const int ka = k0 + 2 * half;
          v2f ar, bu;
          ar.x = Wr[(m * 16 + row) * 32 + ka];
          ar.y = Wr[(m * 16 + row) * 32 + ka + 1];
          bu.x = Ul[(ka + 0) * 16 + row];
          bu.y = Ul[(ka + 1) * 16 + row];
          ra = wmma_k4(ar, bu, ra);
        }
        #pragma unroll
        for (int r = 0; r < 8; ++r) {
          const int c = m * 16 + r + 8 * half;
          Vl[c * 16 + row] = ra[r] + Ul[c * 16 + row];
        }
      }
    }
  } else {
    // ---- Waves 1..7: warm the caches while wave 0 runs the recurrence. ----
    // 128B (32-float) stride per prefetch; recurrence weights first, then the
    // skip/end1 weights needed by phases 2-3.
    const int pid = (wave - 1) * 32 + lane;  // 0..223
    const int NP  = 224;
    #pragma unroll 1
    for (int off = pid; off < (40 * 32 * 32 * 2) / 32; off += NP) {
      __builtin_prefetch(filter_w + off * 32, 0, 1);
      __builtin_prefetch(gate_w   + off * 32, 0, 1);
    }
    #pragma unroll 1
    for (int off = pid; off < (40 * 32 * 32) / 32; off += NP)
      __builtin_prefetch(res_w + off * 32, 0, 1);
    #pragma unroll 1
    for (int off = pid; off < (40 * 256 * 32) / 32; off += NP)
      __builtin_prefetch(skip_w + off * 32, 0, 1);
    #pragma unroll 1
    for (int off = pid; off < (256 * 256) / 32; off += NP)
      __builtin_prefetch(end1_w + off * 32, 0, 1);
  }
  __syncthreads();

  // ---- Phase 2: S = relu( sum_i skip_w[i] @ U_i )  (256 x 8; 2 M-tiles/wave) ----
  #pragma unroll 1
  for (int mt = wave; mt < 16; mt += 8) {
    v8f acc = {};
    #pragma unroll 1
    for (int i = 0; i < NL; ++i) {
      const float* Ws = skip_w + i * (256 * 32);
      const float* Ub = Uh + i * (32 * 8);
      #pragma unroll
      for (int k0 = 0; k0 < 32; k0 += 4) {
        const int ka = k0 + 2 * half;
        v2f a, b;
        a.x = Ws[(mt * 16 + row) * 32 + ka];
        a.y = Ws[(mt * 16 + row) * 32 + ka + 1];
        b.x = (row < 8) ? Ub[(ka + 0) * 8 + row] : 0.0f;  // select, EXEC untouched
        b.y = (row < 8) ? Ub[(ka + 1) * 8 + row] : 0.0f;
        acc = wmma_k4(a, b, acc);
      }
    }
    #pragma unroll
    for (int r = 0; r < 8; ++r) {
      const int c = mt * 16 + r + 8 * half;
      if (row < 8) Sl[c * 8 + row] = fmaxf(acc[r], 0.0f);
    }
  }
  __syncthreads();

  // ---- Phase 3: H = relu( end1_w @ S + end1_b )  (256 x 8, K = 256) ----
  #pragma unroll 1
  for (int mt = wave; mt < 16; mt += 8) {
    v8f acc = {};
    #pragma unroll 8
    for (int k0 = 0; k0 < 256; k0 += 4) {
      const int ka = k0 + 2 * half;
      v2f a, b;
      a.x = end1_w[(mt * 16 + row) * 256 + ka];
      a.y = end1_w[(mt * 16 + row) * 256 + ka + 1];
      b.x = (row < 8) ? Sl[(ka + 0) * 8 + row] : 0.0f;
      b.y = (row < 8) ? Sl[(ka + 1) * 8 + row] : 0.0f;
      acc = wmma_k4(a, b, acc);
    }
    #pragma unroll
    for (int r = 0; r < 8; ++r) {
      const int c = mt * 16 + r + 8 * half;
      if (row < 8) Hl[c * 8 + row] = fmaxf(acc[r] + end1_b[c], 0.0f);
    }
  }
  __syncthreads();

  // ---- Phase 4: out[b] = end2_b + end2_w . H[:, b]  (one batch per wave) ----
  {
    const int b = wave;  // 8 waves == 8 batches
    float s = 0.0f;
    #pragma unroll
    for (int c = lane; c < 256; c += 32)
      s += end2_w[c] * Hl[c * 8 + b];
    #pragma unroll
    for (int off = 16; off > 0; off >>= 1)
      s += __shfl_xor(s, off, 32);
    if (lane == 0) out[b] = s + end2_b[0];
  }
}

extern "C" void kernel_launch(void* const* d_in, const int* in_sizes, int n_in,
                              void* d_out, int out_size, void* d_ws, size_t ws_size,
                              hipStream_t stream) {
  (void)in_sizes; (void)n_in; (void)d_ws; (void)ws_size; (void)out_size;
  const float* x        = (const float*)d_in[0];
  const float* start_w  = (const float*)d_in[1];
  const float* filter_w = (const float*)d_in[2];
  const float* gate_w   = (const float*)d_in[3];
  const float* res_w    = (const float*)d_in[4];
  const float* skip_w   = (const float*)d_in[5];
  const float* end1_w   = (const float*)d_in[6];
  const float* end1_b   = (const float*)d_in[7];
  const float* end2_w   = (const float*)d_in[8];
  const float* end2_b   = (const float*)d_in[9];
  float* out = (float*)d_out;

  wavenet_last_kernel<<<1, 256, 0, stream>>>(x, start_w, filter_w, gate_w, res_w,
                                             skip_w, end1_w, end1_b, end2_w,
                                             end2_b, out);
}